// arneodobouros_67405216743851
// MI455X (gfx1250) — compile-verified
//
#include <hip/hip_runtime.h>
#include <math.h>

// ---------------------------------------------------------------------------
// Problem constants (from reference: D_DATA=1, D_MODEL=ED=4, D_STATE=16,
// DT_RANK=1, D_CONV=4, N_LAYERS=2, B=16, L=16384, dt=1e-4 -> smooth_len=10)
// ---------------------------------------------------------------------------
#define BATCH   16
#define LFULL   16384
#define LC      16376            // L - 8
#define CH      1024             // scan chunk length
#define NC      16               // ceil(LC / CH)
#define NSTACK  3
#define NLAYER  2
#define SMOOTH  10               // int(round(0.001/1e-4))

#define YHAT_N   (BATCH*LC)                // 262016
#define XIN_SZ   (BATCH*LC*4)              // 1048064 floats
#define CTRL_N   (NSTACK*BATCH*LC*4)       // 3144192 floats
#define SCAN_SZ  (NSTACK*BATCH*NC*64)      // 49152 floats

// workspace layout (float offsets)
#define WS_XIN   0
#define WS_ACT   (WS_XIN + XIN_SZ)
#define WS_P     (WS_ACT + NSTACK*XIN_SZ)
#define WS_HL    (WS_P  + SCAN_SZ)
#define WS_HS    (WS_HL + SCAN_SZ)
#define WS_ABSB  (WS_HS + SCAN_SZ)
#define WS_ABSD  (WS_ABSB + YHAT_N)

typedef __attribute__((ext_vector_type(2))) float v2f;
typedef __attribute__((ext_vector_type(8))) float v8f;

struct LayerParams {
  const float* norm_w[NSTACK];
  const float* W_in[NSTACK];     // (8,4) row-major
  const float* conv_w[NSTACK];   // (4,1,4)
  const float* conv_b[NSTACK];   // (4)
  const float* W_x[NSTACK];      // (33,4)
  const float* W_dt[NSTACK];     // (4,1)
  const float* b_dt[NSTACK];     // (4)
  const float* A_log[NSTACK];    // (4,16)
  const float* Dp[NSTACK];       // (4)
  const float* W_out[NSTACK];    // (4,4)
};

// Fast transcendentals: avoid the IEEE v_div_scale/newton divide macro on the
// scan critical path -- single v_rcp_f32 / v_exp_f32 / v_log_f32 each.
__device__ __forceinline__ float rcp_(float x) { return __builtin_amdgcn_rcpf(x); }
__device__ __forceinline__ float silu_(float x) {
  return x * rcp_(1.0f + __expf(-x));
}
__device__ __forceinline__ float softplus_(float x) {
  // stable: softplus(x) = max(x,0) + log(1 + exp(-|x|))
  return fmaxf(x, 0.0f) + __logf(1.0f + __expf(-fabsf(x)));
}

// V_WMMA_F32_16X16X4_F32: D(16x16,f32) = A(16x4,f32) x B(4x16,f32) + C.
// A layout (ISA 7.12.2, 32-bit A 16x4): lane l -> M = l&15, holds K = 2*(l>>4)+{0,1}.
// B layout (symmetric): lane l -> N = l&15, holds K = 2*(l>>4)+{0,1}.
// C/D: VGPR j of lane l -> element [M = j + 8*(l>>4)][N = l&15].
__device__ __forceinline__ v8f wmma4(v2f a, v2f b, v8f c) {
  return __builtin_amdgcn_wmma_f32_16x16x4_f32(false, a, false, b, (short)0, c,
                                               false, false);
}

// ---------------------------------------------------------------------------
// Preprocess: z0 = x[:,4:-4,0], z1 = 9-tap FD derivative, x_in = [z, flip(z)]
// ---------------------------------------------------------------------------
__global__ __launch_bounds__(256)
void prep_kernel(const float* __restrict__ x, float* __restrict__ xin) {
  int i = blockIdx.x * blockDim.x + threadIdx.x;
  if (i >= BATCH * LC) return;
  int b = i / LC, t = i % LC;
  const float* xb = x + (long)b * LFULL;
  const float F0 = 1.0f/280.0f, F1 = -4.0f/105.0f, F2 = 0.2f, F3 = -0.8f,
              F5 = 0.8f, F6 = -0.2f, F7 = 4.0f/105.0f, F8 = -1.0f/280.0f;
  float z0 = xb[t + 4];
  float z1 = F0*xb[t] + F1*xb[t+1] + F2*xb[t+2] + F3*xb[t+3]
           + F5*xb[t+5] + F6*xb[t+6] + F7*xb[t+7] + F8*xb[t+8];
  int r = LC - 1 - t;
  float z0r = xb[r + 4];
  float z1r = F0*xb[r] + F1*xb[r+1] + F2*xb[r+2] + F3*xb[r+3]
            + F5*xb[r+5] + F6*xb[r+6] + F7*xb[r+7] + F8*xb[r+8];
  float* o = xin + (long)i * 4;
  o[0] = z0; o[1] = z1; o[2] = z0r; o[3] = z1r;
}

// ---------------------------------------------------------------------------
// Mamba layer, chunked scan. One wave32 per (chunk, batch, stack).
// PASS 0: compute per-chunk cumulative dA product + local end state.
// PASS 1: run with correct chunk-entry state, write residual output.
// ---------------------------------------------------------------------------
template <int PASS>
__global__ __launch_bounds__(32)
void mamba_layer_kernel(const float* __restrict__ xin_base,
                        float* __restrict__ xout_base,
                        long xin_sstride, long xout_sstride,
                        LayerParams P,
                        float* __restrict__ Pprod,
                        float* __restrict__ Hloc,
                        const float* __restrict__ Hstart) {
  const int c = blockIdx.x, b = blockIdx.y, s = blockIdx.z;
  const int lane = threadIdx.x;
  const int n    = lane & 15;        // state column / WMMA N index
  const int half = lane >> 4;        // K half for WMMA operands
  const int kb   = half * 2;
  const int d_lo = half;             // this lane owns states (d_lo,n),(d_hi,n)
  const int d_hi = 2 + half;

  const float* xin  = xin_base  + (long)s * xin_sstride  + (long)b * LC * 4;
  float*       xout = xout_base + (long)s * xout_sstride + (long)b * LC * 4;

  const float* W_in = P.W_in[s];
  float nw[4], cb[4], cw[16], dpv_lo, dpv_hi;
  #pragma unroll
  for (int j = 0; j < 4;  ++j) { nw[j] = P.norm_w[s][j]; cb[j] = P.conv_b[s][j]; }
  #pragma unroll
  for (int j = 0; j < 16; ++j) cw[j] = P.conv_w[s][j];
  const float wdt_lo = P.W_dt[s][d_lo], wdt_hi = P.W_dt[s][d_hi];
  const float bdt_lo = P.b_dt[s][d_lo], bdt_hi = P.b_dt[s][d_hi];
  const float Aneg_lo = -__expf(P.A_log[s][d_lo*16 + n]);
  const float Aneg_hi = -__expf(P.A_log[s][d_hi*16 + n]);
  dpv_lo = P.Dp[s][d_lo]; dpv_hi = P.Dp[s][d_hi];

  // Stationary B operands (WMMA layout), zero-padded past real columns.
  v2f b_in, b_out, b_x[3];
  b_in.x  = (n < 8) ? W_in[n*4 + kb]     : 0.0f;
  b_in.y  = (n < 8) ? W_in[n*4 + kb + 1] : 0.0f;
  b_out.x = (n < 4) ? P.W_out[s][n*4 + kb]     : 0.0f;
  b_out.y = (n < 4) ? P.W_out[s][n*4 + kb + 1] : 0.0f;
  #pragma unroll
  for (int bt = 0; bt < 3; ++bt) {
    int r = bt * 16 + n;                 // dBC feature row: 0=dlt, 1..16=B, 17..32=C
    b_x[bt].x = (r < 33) ? P.W_x[s][r*4 + kb]     : 0.0f;
    b_x[bt].y = (r < 33) ? P.W_x[s][r*4 + kb + 1] : 0.0f;
  }

  __shared__ float xn[16][4];     // rms-normed input tile (time-major)
  __shared__ float xres[16][4];   // raw input for residual
  __shared__ float xz[16][16];    // input-proj WMMA result (cols 0..7 valid)
  __shared__ float xbuf[16][4];   // silu(conv(xz[:, :4]))
  __shared__ float zbuf[16][4];   // silu(xz[:, 4:8])
  __shared__ float dbc[16][48];   // dBC proj result
  __shared__ float util[16][4];   // u = y * silu(z)
  __shared__ float outD[16][16];  // output-proj WMMA result
  __shared__ float hist[3][4];    // conv ring: last 3 pre-conv rows

  // Conv history at chunk entry: recompute projection of the 3 preceding steps.
  if (lane < 3) {
    int t = c * CH - 3 + lane;
    float h[4] = {0.f, 0.f, 0.f, 0.f};
    if (t >= 0) {
      const float* p = xin + (long)t * 4;
      float v0 = p[0], v1 = p[1], v2 = p[2], v3 = p[3];
      float inv = rsqrtf(0.25f*(v0*v0 + v1*v1 + v2*v2 + v3*v3) + 1e-5f);
      float x0 = v0*inv*nw[0], x1 = v1*inv*nw[1], x2 = v2*inv*nw[2], x3 = v3*inv*nw[3];
      #pragma unroll
      for (int j = 0; j < 4; ++j)
        h[j] = x0*W_in[j*4] + x1*W_in[j*4+1] + x2*W_in[j*4+2] + x3*W_in[j*4+3];
    }
    #pragma unroll
    for (int j = 0; j < 4; ++j) hist[lane][j] = h[j];
  }

  // Scan state: 2 states per lane.
  float h_lo, h_hi, p_lo = 1.0f, p_hi = 1.0f;
  const long sbc = ((long)(s * BATCH + b) * NC + c) * 64;
  if (PASS == 1) { h_lo = Hstart[sbc + lane]; h_hi = Hstart[sbc + 32 + lane]; }
  else           { h_lo = 0.0f;               h_hi = 0.0f; }

  const int t0base = c * CH;
  for (int ti = 0; ti < CH / 16; ++ti) {
    const int t0 = t0base + ti * 16;
    if (t0 >= LC) break;                       // uniform across the wave
    __syncthreads();

    // --- load + RMSNorm (one timestep per lane 0..15) ---
    if (lane < 16) {
      int t = t0 + lane;
      float v0 = 0.f, v1 = 0.f, v2 = 0.f, v3 = 0.f;
      if (t < LC) {
        const float* p = xin + (long)t * 4;
        v0 = p[0]; v1 = p[1]; v2 = p[2]; v3 = p[3];
        if (t + 16 < LC) __builtin_prefetch(xin + (long)(t + 16) * 4, 0, 0);
      }
      xres[lane][0] = v0; xres[lane][1] = v1; xres[lane][2] = v2; xres[lane][3] = v3;
      float inv = rsqrtf(0.25f*(v0*v0 + v1*v1 + v2*v2 + v3*v3) + 1e-5f);
      xn[lane][0] = v0*inv*nw[0]; xn[lane][1] = v1*inv*nw[1];
      xn[lane][2] = v2*inv*nw[2]; xn[lane][3] = v3*inv*nw[3];
    }
    __syncthreads();

    // --- input projection: xz(16x8) = xn(16x4) @ W_in^T via 16x16x4 WMMA ---
    v2f a; a.x = xn[n][kb]; a.y = xn[n][kb + 1];
    v8f dz = {};
    dz = wmma4(a, b_in, dz);
    #pragma unroll
    for (int j = 0; j < 8; ++j) xz[j + 8*half][n] = dz[j];
    __syncthreads();

    // --- causal depthwise conv (K=4) + SiLU, and SiLU of gate ---
    if (lane < 16) {
      int m = lane;
      #pragma unroll
      for (int j = 0; j < 4; ++j) {
        float acc = cb[j];
        #pragma unroll
        for (int q = 0; q < 4; ++q) {
          int r = m - 3 + q;
          float v = (r >= 0) ? xz[r][j] : hist[3 + r][j];
          acc += cw[j*4 + q] * v;
        }
        xbuf[m][j] = silu_(acc);
        zbuf[m][j] = silu_(xz[m][4 + j]);
      }
    }
    __syncthreads();

    // --- dBC projection: (16x33, padded to 48) = xb(16x4) @ W_x^T, 3 WMMAs ---
    v2f ax; ax.x = xbuf[n][kb]; ax.y = xbuf[n][kb + 1];
    #pragma unroll
    for (int bt = 0; bt < 3; ++bt) {
      v8f dd = {};
      dd = wmma4(ax, b_x[bt], dd);
      #pragma unroll
      for (int j = 0; j < 8; ++j) dbc[j + 8*half][bt*16 + n] = dd[j];
    }
    __syncthreads();
    if (lane < 3) {                            // roll conv history forward
      #pragma unroll
      for (int j = 0; j < 4; ++j) hist[lane][j] = xz[13 + lane][j];
    }
    __syncthreads();

    // --- selective scan over the 16 timesteps of this tile (fully unrolled:
    //     LDS offsets become immediates, TRANS ops pipeline across steps) ---
    #pragma unroll
    for (int tt = 0; tt < 16; ++tt) {
      float dlt   = dbc[tt][0];
      float Bn    = dbc[tt][1 + n];
      float xb_lo = xbuf[tt][d_lo], xb_hi = xbuf[tt][d_hi];
      float de_lo = softplus_(dlt * wdt_lo + bdt_lo);
      float de_hi = softplus_(dlt * wdt_hi + bdt_hi);
      float dA_lo = __expf(de_lo * Aneg_lo);
      float dA_hi = __expf(de_hi * Aneg_hi);
      h_lo = dA_lo * h_lo + de_lo * Bn * xb_lo;
      h_hi = dA_hi * h_hi + de_hi * Bn * xb_hi;
      if (PASS == 0) { p_lo *= dA_lo; p_hi *= dA_hi; }
      if (PASS == 1) {
        float Cn   = dbc[tt][17 + n];
        float y_lo = h_lo * Cn, y_hi = h_hi * Cn;
        #pragma unroll
        for (int m = 1; m < 16; m <<= 1) {      // reduce over n within 16-lane group
          y_lo += __shfl_xor(y_lo, m, 32);
          y_hi += __shfl_xor(y_hi, m, 32);
        }
        if (n == 0) {
          util[tt][d_lo] = (y_lo + dpv_lo * xb_lo) * zbuf[tt][d_lo];
          util[tt][d_hi] = (y_hi + dpv_hi * xb_hi) * zbuf[tt][d_hi];
        }
      }
    }

    if (PASS == 1) {
      __syncthreads();
      // --- output projection + residual: out = x + u(16x4) @ W_out^T ---
      v2f au; au.x = util[n][kb]; au.y = util[n][kb + 1];
      v8f dor = {};
      dor = wmma4(au, b_out, dor);
      #pragma unroll
      for (int j = 0; j < 8; ++j) outD[j + 8*half][n] = dor[j];
      __syncthreads();
      if (lane < 16) {
        int t = t0 + lane;
        if (t < LC) {
          float* o = xout + (long)t * 4;
          #pragma unroll
          for (int j = 0; j < 4; ++j) o[j] = xres[lane][j] + outD[lane][j];
        }
      }
    }
  }

  if (PASS == 0) {
    Pprod[sbc + lane]      = p_lo;  Pprod[sbc + 32 + lane] = p_hi;
    Hloc [sbc + lane]      = h_lo;  Hloc [sbc + 32 + lane] = h_hi;
  }
}

// Sequential cross-chunk combine: h_start[c] = P[c-1]*h_start[c-1] + h_loc[c-1]
__global__ __launch_bounds__(64)
void scan_combine_kernel(const float* __restrict__ Pprod,
                         const float* __restrict__ Hloc,
                         float* __restrict__ Hstart) {
  int sb = blockIdx.x;          // 0 .. NSTACK*BATCH-1
  int st = threadIdx.x;         // 0 .. 63 (state id)
  float carry = 0.0f;
  for (int c = 0; c < NC; ++c) {
    long k = ((long)sb * NC + c) * 64 + st;
    Hstart[k] = carry;
    carry = Pprod[k] * carry + Hloc[k];
  }
}

// ---------------------------------------------------------------------------
// Heads: |beta|, |d| then smoothing + final polynomial + zeros scalar.
// controls layout in d_out: rows 0-15 alpha, 16-31 beta, 32-47 d.
// ---------------------------------------------------------------------------
__global__ __launch_bounds__(256)
void head_abs_kernel(const float* __restrict__ ctrl,
                     const float* __restrict__ bw, const float* __restrict__ bb,
                     const float* __restrict__ dw, const float* __restrict__ db,
                     float* __restrict__ absb, float* __restrict__ absd) {
  int i = blockIdx.x * blockDim.x + threadIdx.x;
  if (i >= BATCH * LC) return;
  int b = i / LC, t = i % LC;
  const float* bC = ctrl + (((long)(BATCH   + b) * LC) + t) * 4;
  const float* dC = ctrl + (((long)(2*BATCH + b) * LC) + t) * 4;
  float bet = bC[0]*bw[0] + bC[1]*bw[1] + bC[2]*bw[2] + bC[3]*bw[3] + bb[0];
  float dd  = dC[0]*dw[0] + dC[1]*dw[1] + dC[2]*dw[2] + dC[3]*dw[3] + db[0];
  absb[i] = fabsf(bet);
  absd[i] = fabsf(dd);
}

__global__ __launch_bounds__(256)
void head_final_kernel(const float* __restrict__ ctrl,
                       const float* __restrict__ xin,
                       const float* __restrict__ aw, const float* __restrict__ ab,
                       const float* __restrict__ absb, const float* __restrict__ absd,
                       const float* __restrict__ dtp, const float* __restrict__ taup,
                       float* __restrict__ yhat, float* __restrict__ zscalar) {
  int i = blockIdx.x * blockDim.x + threadIdx.x;
  if (i == 0) zscalar[0] = 0.0f;
  if (i >= BATCH * LC) return;
  int b = i / LC, t = i % LC;
  const float* aC = ctrl + (((long)b * LC) + t) * 4;
  float alpha = aC[0]*aw[0] + aC[1]*aw[1] + aC[2]*aw[2] + aC[3]*aw[3] + ab[0];
  float sb = 0.f, sd = 0.f;                         // SAME conv, pad (4,5), /10
  #pragma unroll
  for (int j = -4; j <= 5; ++j) {
    int tt = t + j;
    if (tt >= 0 && tt < LC) { sb += absb[(long)b*LC + tt]; sd += absd[(long)b*LC + tt]; }
  }
  sb *= (1.0f / SMOOTH); sd *= (1.0f / SMOOTH);
  const float* xi = xin + (long)i * 4;
  float z1 = xi[0];
  float z2 = xi[1] / dtp[0];
  float tau = taup[0];
  float z1_2 = z1 * z1, z1_3 = z1_2 * z1;
  yhat[i] = alpha + sb*sb*z1 + z1_2 - z1_3 - z1*z2/tau - z1_2*z2/tau - sd;
}

// ---------------------------------------------------------------------------
// Host-side launch. Input flattening assumed in setup_inputs() insertion
// order: x, dt, then 3 stacks x 2 layers x {norm_w,W_in,conv_w,conv_b,W_x,
// W_dt,b_dt,A_log,Dp,W_out}, then alpha_w, alpha_b, beta_w, beta_b, d_w,
// d_b, tau  (n_in = 69).
// d_out = [yhat (262016) | controls (3144192) | zeros (1)].
// ---------------------------------------------------------------------------
extern "C" void kernel_launch(void* const* d_in, const int* in_sizes, int n_in,
                              void* d_out, int out_size, void* d_ws, size_t ws_size,
                              hipStream_t stream) {
  (void)in_sizes; (void)out_size; (void)ws_size;
  if (n_in < 69) return;

  const float* x    = (const float*)d_in[0];
  const float* dtp  = (const float*)d_in[1];
  auto blk = [&](int s, int layer, int j) -> const float* {
    return (const float*)d_in[2 + s*20 + layer*10 + j];
  };
  const float* alpha_w = (const float*)d_in[62];
  const float* alpha_b = (const float*)d_in[63];
  const float* beta_w  = (const float*)d_in[64];
  const float* beta_b  = (const float*)d_in[65];
  const float* d_w     = (const float*)d_in[66];
  const float* d_b     = (const float*)d_in[67];
  const float* tau     = (const float*)d_in[68];

  float* ws   = (float*)d_ws;
  float* XIN  = ws + WS_XIN;
  float* ACT  = ws + WS_ACT;
  float* Pbuf = ws + WS_P;
  float* Hl   = ws + WS_HL;
  float* Hs   = ws + WS_HS;
  float* ABSB = ws + WS_ABSB;
  float* ABSD = ws + WS_ABSD;

  float* yhat = (float*)d_out;
  float* ctrl = yhat + YHAT_N;
  float* zsc  = ctrl + CTRL_N;

  const int NTH = 256;
  const int NBK = (BATCH * LC + NTH - 1) / NTH;

  prep_kernel<<<NBK, NTH, 0, stream>>>(x, XIN);

  for (int layer = 0; layer < NLAYER; ++layer) {
    LayerParams P;
    for (int s = 0; s < NSTACK; ++s) {
      P.norm_w[s] = blk(s, layer, 0);  P.W_in[s]  = blk(s, layer, 1);
      P.conv_w[s] = blk(s, layer, 2);  P.conv_b[s]= blk(s, layer, 3);
      P.W_x[s]    = blk(s, layer, 4);  P.W_dt[s]  = blk(s, layer, 5);
      P.b_dt[s]   = blk(s, layer, 6);  P.A_log[s] = blk(s, layer, 7);
      P.Dp[s]     = blk(s, layer, 8);  P.W_out[s] = blk(s, layer, 9);
    }
    const float* lin = (layer == 0) ? XIN : ACT;
    long lin_stride  = (layer == 0) ? 0 : (long)XIN_SZ;  // layer0 input shared
    float* lout      = (layer == 0) ? ACT : ctrl;        // layer1 -> controls
    long lout_stride = (long)XIN_SZ;

    dim3 g(NC, BATCH, NSTACK);
    mamba_layer_kernel<0><<<g, 32, 0, stream>>>(lin, lout, lin_stride, lout_stride,
                                                P, Pbuf, Hl, Hs);
    scan_combine_kernel<<<NSTACK * BATCH, 64, 0, stream>>>(Pbuf, Hl, Hs);
    mamba_layer_kernel<1><<<g, 32, 0, stream>>>(lin, lout, lin_stride, lout_stride,
                                                P, Pbuf, Hl, Hs);
  }

  head_abs_kernel<<<NBK, NTH, 0, stream>>>(ctrl, beta_w, beta_b, d_w, d_b, ABSB, ABSD);
  head_final_kernel<<<NBK, NTH, 0, stream>>>(ctrl, XIN, alpha_w, alpha_b,
                                             ABSB, ABSD, dtp, tau, yhat, zsc);
}